// GraphLaplacianAttention_57208964383548
// MI455X (gfx1250) — compile-verified
//
#include <hip/hip_runtime.h>

#define DIM 64
#define EDIM 32
#define NHEADS 8
#define EXPH 16
#define HDIM 8
#define SCALE 0.35355339059327373f /* 8^-0.5 */
#define EPSV 1e-16f

typedef float v2f __attribute__((ext_vector_type(2)));
typedef float v8f __attribute__((ext_vector_type(8)));
typedef unsigned int u32x4 __attribute__((ext_vector_type(4)));
typedef int i32x4 __attribute__((ext_vector_type(4)));
typedef int i32x8 __attribute__((ext_vector_type(8)));

__device__ __forceinline__ v8f wmma4(v2f a, v2f b, v8f c) {
  // D = A(16x4,f32) * B(4x16,f32) + C(16x16,f32)
  return __builtin_amdgcn_wmma_f32_16x16x4_f32(false, a, false, b, (short)0, c,
                                               false, false);
}

// ---- Tensor Data Mover: 1-row DMA of `tile_dw` dwords global -> LDS --------
// tensor_dim0 = tensor_dw (reads past it return zero), tile_dim0 = tile_dw.
__device__ __forceinline__ void tdm_load_dwords(void* lds_ptr, const void* gptr,
                                                unsigned tile_dw,
                                                unsigned tensor_dw) {
  const unsigned lds_off =
      __builtin_amdgcn_readfirstlane((unsigned)(uintptr_t)lds_ptr);
  const size_t ga = (size_t)gptr;
  const unsigned galo = __builtin_amdgcn_readfirstlane((unsigned)ga);
  const unsigned gahi = __builtin_amdgcn_readfirstlane((unsigned)(ga >> 32));
  const unsigned td = __builtin_amdgcn_readfirstlane(tensor_dw);

  u32x4 g0;
  g0[0] = 1u;                                 // count=1 (valid), user mode
  g0[1] = lds_off;                            // lds_addr (bytes)
  g0[2] = galo;                               // global_addr[31:0]
  g0[3] = (gahi & 0x01FFFFFFu) | (2u << 30);  // global_addr[56:32] | type=2

  i32x8 g1;
  g1[0] = (int)(2u << 16);                  // data_size=4B; wg_mask=0; no flags
  g1[1] = (int)((td & 0xFFFFu) << 16);      // tensor_dim0[15:0]
  g1[2] = (int)(((td >> 16) & 0xFFFFu) | (1u << 16));  // dim0[31:16], dim1=1
  g1[3] = (int)((tile_dw & 0xFFFFu) << 16); // tile_dim0
  g1[4] = 1;                                // tile_dim1=1, tile_dim2=0
  g1[5] = (int)tile_dw;                     // tensor_dim0_stride lo32
  g1[6] = 0;
  g1[7] = 0;

  i32x4 z4 = {0, 0, 0, 0};
#if __clang_major__ >= 23
  i32x8 z8 = {0, 0, 0, 0, 0, 0, 0, 0};
  __builtin_amdgcn_tensor_load_to_lds(g0, g1, z4, z4, z8, 0);
#else
  __builtin_amdgcn_tensor_load_to_lds(g0, g1, z4, z4, 0);
#endif
}

__device__ __forceinline__ void tdm_wait() {
  __builtin_amdgcn_s_wait_tensorcnt(0);
  asm volatile("" ::: "memory");  // TDM wrote LDS behind the compiler's back
}

// monotone order-preserving f32 <-> u32 encoding for atomic segment-max
__device__ __forceinline__ unsigned encf(float f) {
  unsigned u = __float_as_uint(f);
  return (u & 0x80000000u) ? ~u : (u | 0x80000000u);
}
__device__ __forceinline__ float decf(unsigned u) {
  return (u & 0x80000000u) ? __uint_as_float(u & 0x7FFFFFFFu)
                           : __uint_as_float(~u);
}

// ---------------- init scratch (segmax encoded-zero, segsum, agg) -----------
__global__ void init_ws(unsigned* __restrict__ segmax, float* __restrict__ segsum,
                        float* __restrict__ agg, int n) {
  int t = blockIdx.x * blockDim.x + threadIdx.x;
  if (t < n * EXPH) { segmax[t] = 0u; segsum[t] = 0.f; }
  if (t < n * DIM) agg[t] = 0.f;
}

// 16x64 = (16xK=64 LDS tile) @ (64x64 W) via 4 col-tiles x 16 k-steps of WMMA.
// Separate inlined call per weight keeps W in addrspace(1) -> global_load.
__device__ __forceinline__ void proj_tile16(const float* xsw,
                                            const float* __restrict__ W,
                                            float* __restrict__ O, int row0,
                                            int n, bool live, int m, int kb,
                                            int mb) {
  for (int t = 0; t < 4; ++t) {
    const int nn = (t << 4) + m;
    v8f acc = {};
#pragma unroll
    for (int kk = 0; kk < 16; ++kk) {
      const int kbase = (kk << 2) + kb;
      v2f a, b;
      a[0] = xsw[m * DIM + kbase];
      a[1] = xsw[m * DIM + kbase + 1];
      b[0] = W[(size_t)kbase * DIM + nn];
      b[1] = W[(size_t)(kbase + 1) * DIM + nn];
      acc = wmma4(a, b, acc);
    }
    if (live) {
      if (row0 + 16 <= n) {
#pragma unroll
        for (int j = 0; j < 8; ++j)
          O[(size_t)(row0 + mb + j) * DIM + nn] = acc[j];
      } else {
#pragma unroll
        for (int j = 0; j < 8; ++j) {
          const int r = row0 + mb + j;
          if (r < n) O[(size_t)r * DIM + nn] = acc[j];
        }
      }
    }
  }
}

// ---------------- q,k,v = x @ {Wq,Wk,Wv} via WMMA f32 16x16x4 ---------------
__global__ __launch_bounds__(128)
void qkv_proj(const float* __restrict__ x, const float* __restrict__ Wq,
              const float* __restrict__ Wk, const float* __restrict__ Wv,
              float* __restrict__ q, float* __restrict__ k,
              float* __restrict__ v, int n) {
  __shared__ float xs[4][16 * DIM];
  const int wave = threadIdx.x >> 5, lane = threadIdx.x & 31;
  const int ntiles = (n + 15) >> 4;
  int tile = blockIdx.x * 4 + wave;
  const bool live = tile < ntiles;
  if (!live) tile = 0;
  const int row0 = tile << 4;

  // DMA the contiguous 16x64 f32 x-tile into this wave's LDS slice.
  tdm_load_dwords(&xs[wave][0], x + (size_t)row0 * DIM, 16 * DIM,
                  (unsigned)(n - row0) * DIM);
  tdm_wait();
  __syncthreads();

  const int m = lane & 15, kb = (lane >> 4) << 1, mb = (lane >> 4) << 3;
  proj_tile16(xs[wave], Wq, q, row0, n, live, m, kb, mb);
  proj_tile16(xs[wave], Wk, k, row0, n, live, m, kb, mb);
  proj_tile16(xs[wave], Wv, v, row0, n, live, m, kb, mb);
}

// 32x64 edge-projection tile: (32xK=32 LDS edges) @ (32x64 W) -> LDS out.
__device__ __forceinline__ void edge_proj_tile32(const float* esw,
                                                 const float* __restrict__ W,
                                                 float* outw, int m, int kb,
                                                 int mb) {
  for (int sub = 0; sub < 2; ++sub) {
    for (int t = 0; t < 4; ++t) {
      const int nn = (t << 4) + m;
      v8f acc = {};
#pragma unroll
      for (int kk = 0; kk < 8; ++kk) {
        const int kbase = (kk << 2) + kb;
        v2f a, b;
        a[0] = esw[(sub * 16 + m) * EDIM + kbase];
        a[1] = esw[(sub * 16 + m) * EDIM + kbase + 1];
        b[0] = W[(size_t)kbase * DIM + nn];
        b[1] = W[(size_t)(kbase + 1) * DIM + nn];
        acc = wmma4(a, b, acc);
      }
#pragma unroll
      for (int j = 0; j < 8; ++j)
        outw[(sub * 16 + mb + j) * DIM + nn] = acc[j];
    }
  }
}

// ------- edge pass 1: edge_k = edges@Wek (WMMA), logits, @Wexp --------------
__global__ __launch_bounds__(128)
void edge_logits(const float* __restrict__ edges, const int* __restrict__ eidx,
                 const float* __restrict__ q, const float* __restrict__ k,
                 const float* __restrict__ Wek, const float* __restrict__ Wexp,
                 const float* __restrict__ temperature,
                 float* __restrict__ attn_ex, int nE) {
  __shared__ float es[4][32 * EDIM];
  __shared__ float eks[4][32 * DIM];
  const int wave = threadIdx.x >> 5, lane = threadIdx.x & 31;
  const int e0 = (blockIdx.x * 4 + wave) * 32;

  tdm_load_dwords(&es[wave][0], edges + (size_t)e0 * EDIM, 32 * EDIM,
                  (unsigned)(nE > e0 ? (nE - e0) : 0) * EDIM);
  tdm_wait();
  __syncthreads();

  const int m = lane & 15, kb = (lane >> 4) << 1, mb = (lane >> 4) << 3;
  edge_proj_tile32(es[wave], Wek, eks[wave], m, kb, mb);
  __syncthreads();

  const int e = e0 + lane;
  if (e >= nE) return;
  const int dstN = eidx[e];
  const int srcN = eidx[nE + e];
  const float invT = 1.f / temperature[0];
  float logit[NHEADS];
#pragma unroll
  for (int h = 0; h < NHEADS; ++h) {
    float dot = 0.f;
#pragma unroll
    for (int d = 0; d < HDIM; ++d) {
      const int c = h * HDIM + d;
      const float kk2 =
          (k[(size_t)srcN * DIM + c] + eks[wave][lane * DIM + c]) * SCALE;
      dot += q[(size_t)dstN * DIM + c] * kk2;
    }
    logit[h] = dot * invT;
  }
#pragma unroll
  for (int j = 0; j < EXPH; ++j) {
    float s = 0.f;
#pragma unroll
    for (int h = 0; h < NHEADS; ++h) s += logit[h] * Wexp[h * EXPH + j];
    attn_ex[(size_t)e * EXPH + j] = s;
  }
}

// ---------------- segment max (ordered-uint atomicMax) ----------------------
__global__ void seg_max_k(const float* __restrict__ attn_ex,
                          const int* __restrict__ eidx,
                          unsigned* __restrict__ segmax, int nE) {
  const int t = blockIdx.x * blockDim.x + threadIdx.x;
  if (t >= nE * EXPH) return;
  const int e = t >> 4, j = t & 15;
  atomicMax(&segmax[eidx[e] * EXPH + j], encf(attn_ex[t]));
}

// ---------------- exp(x-max) in place + segment sum -------------------------
__global__ void seg_expsum_k(float* __restrict__ attn_ex,
                             const int* __restrict__ eidx,
                             const unsigned* __restrict__ segmax,
                             float* __restrict__ segsum, int nE) {
  const int t = blockIdx.x * blockDim.x + threadIdx.x;
  if (t >= nE * EXPH) return;
  const int e = t >> 4, j = t & 15;
  const int d = eidx[e];
  const float ex = __expf(attn_ex[t] - decf(segmax[d * EXPH + j]));
  attn_ex[t] = ex;
  atomicAdd(&segsum[d * EXPH + j], ex);
}

// ---- edge pass 2: edge_v (WMMA), normalize, @Wsq+bsq, scatter aggregate ----
__global__ __launch_bounds__(128)
void edge_final(const float* __restrict__ edges, const int* __restrict__ eidx,
                const float* __restrict__ v, const float* __restrict__ Wev,
                const float* __restrict__ Wsq, const float* __restrict__ bsq,
                const float* __restrict__ attn_ex,
                const float* __restrict__ segsum, float* __restrict__ agg,
                float* __restrict__ out_attn, int nE) {
  __shared__ float es[4][32 * EDIM];
  __shared__ float evs[4][32 * DIM];
  const int wave = threadIdx.x >> 5, lane = threadIdx.x & 31;
  const int e0 = (blockIdx.x * 4 + wave) * 32;

  tdm_load_dwords(&es[wave][0], edges + (size_t)e0 * EDIM, 32 * EDIM,
                  (unsigned)(nE > e0 ? (nE - e0) : 0) * EDIM);
  tdm_wait();
  __syncthreads();

  const int m = lane & 15, kb = (lane >> 4) << 1, mb = (lane >> 4) << 3;
  edge_proj_tile32(es[wave], Wev, evs[wave], m, kb, mb);
  __syncthreads();

  const int e = e0 + lane;
  if (e >= nE) return;
  const int dstN = eidx[e];
  const int srcN = eidx[nE + e];
  float ah[NHEADS];
#pragma unroll
  for (int h = 0; h < NHEADS; ++h) ah[h] = bsq[h];
#pragma unroll
  for (int j = 0; j < EXPH; ++j) {
    const float a =
        attn_ex[(size_t)e * EXPH + j] / (segsum[dstN * EXPH + j] + EPSV);
#pragma unroll
    for (int h = 0; h < NHEADS; ++h) ah[h] += a * Wsq[j * NHEADS + h];
  }
#pragma unroll
  for (int h = 0; h < NHEADS; ++h) out_attn[(size_t)h * nE + e] = ah[h];
#pragma unroll
  for (int c = 0; c < DIM; ++c) {
    const float vc = v[(size_t)srcN * DIM + c] + evs[wave][lane * DIM + c];
    atomicAdd(&agg[(size_t)dstN * DIM + c], ah[c >> 3] * vc);
  }
}

// ---------------- out = (v - agg) @ Wout + bout -----------------------------
__global__ __launch_bounds__(128)
void out_proj(const float* __restrict__ v, const float* __restrict__ agg,
              const float* __restrict__ Wout, const float* __restrict__ bout,
              float* __restrict__ out, int n) {
  __shared__ float xs[4][16 * DIM];
  const int wave = threadIdx.x >> 5, lane = threadIdx.x & 31;
  const int ntiles = (n + 15) >> 4;
  int tile = blockIdx.x * 4 + wave;
  const bool live = tile < ntiles;
  if (!live) tile = 0;
  const int row0 = tile << 4;
  for (int i = lane; i < 16 * DIM; i += 32) {
    const size_t gi = (size_t)row0 * DIM + i;
    xs[wave][i] = (row0 + (i >> 6) < n) ? (v[gi] - agg[gi]) : 0.f;
  }
  __syncthreads();

  const int m = lane & 15, kb = (lane >> 4) << 1, mb = (lane >> 4) << 3;
  for (int t = 0; t < 4; ++t) {
    const int nn = (t << 4) + m;
    v8f acc = {};
#pragma unroll
    for (int kk = 0; kk < 16; ++kk) {
      const int kbase = (kk << 2) + kb;
      v2f a, b;
      a[0] = xs[wave][m * DIM + kbase];
      a[1] = xs[wave][m * DIM + kbase + 1];
      b[0] = Wout[(size_t)kbase * DIM + nn];
      b[1] = Wout[(size_t)(kbase + 1) * DIM + nn];
      acc = wmma4(a, b, acc);
    }
    if (live) {
      const float bo = bout[nn];
      if (row0 + 16 <= n) {
#pragma unroll
        for (int j = 0; j < 8; ++j)
          out[(size_t)(row0 + mb + j) * DIM + nn] = acc[j] + bo;
      } else {
#pragma unroll
        for (int j = 0; j < 8; ++j) {
          const int r = row0 + mb + j;
          if (r < n) out[(size_t)r * DIM + nn] = acc[j] + bo;
        }
      }
    }
  }
}

extern "C" void kernel_launch(void* const* d_in, const int* in_sizes, int n_in,
                              void* d_out, int out_size, void* d_ws,
                              size_t ws_size, hipStream_t stream) {
  const float* x    = (const float*)d_in[0];
  const float* edges= (const float*)d_in[1];
  const int*   eidx = (const int*)d_in[2];
  const float* Wq   = (const float*)d_in[3];
  const float* Wk   = (const float*)d_in[4];
  const float* Wv   = (const float*)d_in[5];
  const float* Wek  = (const float*)d_in[6];
  const float* Wev  = (const float*)d_in[7];
  const float* Wexp = (const float*)d_in[8];
  const float* Wsq  = (const float*)d_in[9];
  const float* bsq  = (const float*)d_in[10];
  const float* Wout = (const float*)d_in[11];
  const float* bout = (const float*)d_in[12];
  const float* temp = (const float*)d_in[13];

  const int n  = in_sizes[0] / DIM;
  const int nE = in_sizes[1] / EDIM;

  float* ws = (float*)d_ws;
  float* q   = ws;
  float* k   = q + (size_t)n * DIM;
  float* v   = k + (size_t)n * DIM;
  float* agg = v + (size_t)n * DIM;
  unsigned* segmax = (unsigned*)(agg + (size_t)n * DIM);
  float* segsum  = (float*)(segmax + (size_t)n * EXPH);
  float* attn_ex = segsum + (size_t)n * EXPH;

  float* out      = (float*)d_out;
  float* out_attn = out + (size_t)n * DIM;

  const int initN = n * DIM;
  init_ws<<<(initN + 255) / 256, 256, 0, stream>>>(segmax, segsum, agg, n);

  const int ntiles = (n + 15) >> 4;
  const int nblkA  = (ntiles + 3) >> 2;
  qkv_proj<<<nblkA, 128, 0, stream>>>(x, Wq, Wk, Wv, q, k, v, n);

  const int nblkE = (nE + 127) >> 7;
  edge_logits<<<nblkE, 128, 0, stream>>>(edges, eidx, q, k, Wek, Wexp, temp,
                                         attn_ex, nE);

  const int nEl = nE * EXPH;
  seg_max_k<<<(nEl + 255) / 256, 256, 0, stream>>>(attn_ex, eidx, segmax, nE);
  seg_expsum_k<<<(nEl + 255) / 256, 256, 0, stream>>>(attn_ex, eidx, segmax,
                                                      segsum, nE);

  edge_final<<<nblkE, 128, 0, stream>>>(edges, eidx, v, Wev, Wsq, bsq, attn_ex,
                                        segsum, agg, out_attn, nE);

  out_proj<<<nblkA, 128, 0, stream>>>(v, agg, Wout, bout, out, n);
}